// LIDARStateCost_32701880991911
// MI455X (gfx1250) — compile-verified
//
#include <hip/hip_runtime.h>
#include <math.h>

// LIDAR plane-fit state cost.
//   inputs:  xt[N,3] f32, dataset[M,3] f32, idx[N,20] int32
//   output:  out[N] f32
// Memory-bound gather kernel: dataset (24 MB) is L2-resident on MI455X
// (192 MB L2); one thread per point, all 20 gathers issued before use.

#define K_NEI 20
#define LIMF  5.0f

__device__ __forceinline__ float sigmoid_f(float u) {
    return 1.0f / (1.0f + expf(-u));
}

extern "C" __global__ __launch_bounds__(256, 1)
void lidar_cost_kernel(const float* __restrict__ xt,
                       const float* __restrict__ dataset,
                       const int*   __restrict__ idx,
                       float* __restrict__ out,
                       int n, int stride)
{
    int tid = blockIdx.x * blockDim.x + threadIdx.x;

    for (int p = tid; p < n; p += stride) {
        // ---- prefetch next iteration's index row (global_prefetch_b8) ----
        int pnext = p + stride;
        if (pnext < n) {
            const char* nrow = (const char*)(idx + (long long)pnext * K_NEI);
            __builtin_prefetch(nrow, 0, 0);        // row spans 80 bytes:
            __builtin_prefetch(nrow + 64, 0, 0);   // cover both cachelines
        }

        // ---- load 20 indices: 5 x int4 (rows are 80B, 16B-aligned) ----
        const int4* __restrict__ irow = (const int4*)(idx + (long long)p * K_NEI);
        int4 a0 = irow[0];
        int4 a1 = irow[1];
        int4 a2 = irow[2];
        int4 a3 = irow[3];
        int4 a4 = irow[4];
        int id[K_NEI];
        id[ 0]=a0.x; id[ 1]=a0.y; id[ 2]=a0.z; id[ 3]=a0.w;
        id[ 4]=a1.x; id[ 5]=a1.y; id[ 6]=a1.z; id[ 7]=a1.w;
        id[ 8]=a2.x; id[ 9]=a2.y; id[10]=a2.z; id[11]=a2.w;
        id[12]=a3.x; id[13]=a3.y; id[14]=a3.z; id[15]=a3.w;
        id[16]=a4.x; id[17]=a4.y; id[18]=a4.z; id[19]=a4.w;

        // ---- phase 1: issue all 20 gathers (global_load_b96, L2 hits) ----
        float qx[K_NEI], qy[K_NEI], qz[K_NEI];
        #pragma unroll
        for (int k = 0; k < K_NEI; ++k) {
            const float* q = dataset + 3ll * (unsigned)id[k];
            qx[k] = q[0];
            qy[k] = q[1];
            qz[k] = q[2];
        }

        // xt row for this point (12B, coalesced)
        const float* xp = xt + 3ll * p;
        float X = xp[0];
        float Y = xp[1];
        float Z = xp[2];

        // ---- phase 2: accumulate normal equations A = D^T D, b = D^T z ----
        // D row = [x, y, 1];  A is symmetric 3x3, A[2][2] = K.
        float Sxx = 0.f, Sxy = 0.f, Syy = 0.f;
        float Sx  = 0.f, Sy  = 0.f;
        float Sxz = 0.f, Syz = 0.f, Sz = 0.f;
        #pragma unroll
        for (int k = 0; k < K_NEI; ++k) {
            float x = qx[k], y = qy[k], z = qz[k];
            Sxx = fmaf(x, x, Sxx);
            Sxy = fmaf(x, y, Sxy);
            Syy = fmaf(y, y, Syy);
            Sx += x;
            Sy += y;
            Sxz = fmaf(x, z, Sxz);
            Syz = fmaf(y, z, Syz);
            Sz += z;
        }
        const float S1 = (float)K_NEI;

        // ---- 3x3 solve via symmetric adjugate (Cramer) ----
        float c00 = fmaf(Syy, S1, -Sy * Sy);
        float c01 = fmaf(Sx,  Sy, -Sxy * S1);
        float c02 = fmaf(Sxy, Sy, -Syy * Sx);
        float c11 = fmaf(Sxx, S1, -Sx * Sx);
        float c12 = fmaf(Sxy, Sx, -Sxx * Sy);
        float c22 = fmaf(Sxx, Syy, -Sxy * Sxy);
        float det = Sxx * c00 + Sxy * c01 + Sx * c02;
        float inv = 1.0f / det;

        float w0 = (c00 * Sxz + c01 * Syz + c02 * Sz) * inv;
        float w1 = (c01 * Sxz + c11 * Syz + c12 * Sz) * inv;
        float w2 = (c02 * Sxz + c12 * Syz + c22 * Sz) * inv;

        // ---- cost terms ----
        // n = (w0, w1, -1); pn = xt.n; nn = |n|^2; t = (pn + w2)/nn
        float pn = fmaf(X, w0, fmaf(Y, w1, -Z));
        float nn = fmaf(w0, w0, fmaf(w1, w1, 1.0f));
        float t  = (pn + w2) / nn;

        float closeness = t * t * nn;          // |t*n|^2
        float height    = expf(Z + t);         // proj_z = z - t*(-1)
        float boundary  =
            sigmoid_f((X - LIMF) * 10.0f) + 1.0f - sigmoid_f((X + LIMF) * 10.0f) +
            sigmoid_f((Y - LIMF) * 10.0f) + 1.0f - sigmoid_f((Y + LIMF) * 10.0f);

        out[p] = closeness + height + boundary;
    }
}

extern "C" void kernel_launch(void* const* d_in, const int* in_sizes, int n_in,
                              void* d_out, int out_size, void* d_ws, size_t ws_size,
                              hipStream_t stream)
{
    (void)n_in; (void)d_ws; (void)ws_size;
    const float* xt      = (const float*)d_in[0];
    const float* dataset = (const float*)d_in[1];
    const int*   idx     = (const int*)d_in[2];
    float*       out     = (float*)d_out;

    int n = in_sizes[0] / 3;           // N points
    (void)out_size;

    const int block = 256;             // 8 wave32 per block
    const int iters = 4;               // grid-stride depth (enables prefetch)
    int blocks = (n + block * iters - 1) / (block * iters);
    if (blocks < 1) blocks = 1;
    int stride = blocks * block;

    lidar_cost_kernel<<<blocks, block, 0, stream>>>(xt, dataset, idx, out, n, stride);
}